// GCNLayer_53326313947257
// MI455X (gfx1250) — compile-verified
//
#include <hip/hip_runtime.h>
#include <hip/hip_bf16.h>
#include <stdint.h>

// ---------------------------------------------------------------------------
// GCNConv forward for MI455X (gfx1250, wave32).
//   out = scatter_add( (x@W)[row] * dinv[row]*dinv[col] -> col ) + b
// Dominant cost: 3.2M-edge gather/scatter (~6.4 GB, L2-resident) ->
//   wave-per-edge, float4 gathers, hardware global_atomic_add_f32.
// GEMM phase uses V_WMMA_F32_16X16X4_F32 with W repacked into fragment order
// so B fragments are single coalesced b64 loads hoisted in clauses (no
// per-WMMA s_wait_loadcnt 0 stalls).
// ---------------------------------------------------------------------------

typedef float  v2f __attribute__((ext_vector_type(2)));
typedef float  v8f __attribute__((ext_vector_type(8)));

#define D 128          // D_IN == D_OUT == 128
#define NT 8           // column tiles (128/16)
#define KK 32          // K steps (128/4)

// ---------------- K0: repack W into WMMA B-fragment order ------------------
// Wt layout (v2f units): [(nt*32 + kk)*32 + lane] =
//   { W[4kk + 2*(lane>>4)    ][16nt + (lane&15)],
//     W[4kk + 2*(lane>>4) + 1][16nt + (lane&15)] }
__global__ void k_pack_w(const float* __restrict__ W, float* __restrict__ Wt) {
    int t = blockIdx.x * blockDim.x + threadIdx.x;   // 0 .. 8191
    if (t >= NT * KK * 32) return;
    int lane = t & 31;
    int kk   = (t >> 5) & 31;
    int nt   = t >> 10;
    int m    = lane & 15;
    int kp   = lane >> 4;
    int kb   = 4 * kk + 2 * kp;
    int n    = 16 * nt + m;
    Wt[2 * t + 0] = W[(size_t)(kb + 0) * D + n];
    Wt[2 * t + 1] = W[(size_t)(kb + 1) * D + n];
}

// ---------------- K1: init degree with self-loop count --------------------
__global__ void k_init_deg(unsigned* __restrict__ deg, int n) {
    int i = blockIdx.x * blockDim.x + threadIdx.x;
    if (i < n) deg[i] = 1u;
}

// ---------------- K2: XW = X @ W via WMMA f32 16x16x4 ---------------------
// One wave per 16-row strip. A fragments (full K) hoisted once; per column
// tile, all 32 B fragments loaded in one clause from the repacked Wt, then
// 32 back-to-back WMMAs.
__global__ void k_gemm_wmma(const float* __restrict__ X,
                            const float* __restrict__ Wt,
                            float* __restrict__ XW, int nrows) {
    const int wave = (int)((blockIdx.x * (unsigned)blockDim.x + threadIdx.x) >> 5);
    const int lane = threadIdx.x & 31;
    const int row0 = wave * 16;
    if (row0 >= nrows) return;             // wave-uniform: EXEC stays all-1s

    const int m  = lane & 15;
    const int kp = lane >> 4;

    const float* xrow = X + (size_t)(row0 + m) * D;

    // A fragments for K = 0..127 (32 steps of K=4): 64 VGPRs, b64 loads.
    v2f afr[KK];
#pragma unroll
    for (int kk = 0; kk < KK; ++kk) {
        const int kb = 4 * kk + 2 * kp;
        afr[kk].x = xrow[kb + 0];
        afr[kk].y = xrow[kb + 1];
    }

    const v2f* wt = reinterpret_cast<const v2f*>(Wt) + lane;

#pragma unroll
    for (int nt = 0; nt < NT; ++nt) {
        // Hoist all B fragments for this column tile: clause of 32 b64 loads.
        v2f bfr[KK];
#pragma unroll
        for (int kk = 0; kk < KK; ++kk)
            bfr[kk] = wt[(nt * KK + kk) * 32];

        v8f acc = {};
#pragma unroll
        for (int kk = 0; kk < KK; ++kk)
            acc = __builtin_amdgcn_wmma_f32_16x16x4_f32(
                false, afr[kk], false, bfr[kk],
                (short)0, acc, false, false);

        // Store D: VGPR r -> row row0 + 8*kp + r, col 16*nt + m.
        const int mbase = row0 + 8 * kp;
        const int nn = 16 * nt + m;
#pragma unroll
        for (int r = 0; r < 8; ++r)
            XW[(size_t)(mbase + r) * D + nn] = acc[r];
    }
}

// ---------------- K3: target-degree histogram -----------------------------
__global__ void k_degree(const int* __restrict__ col, unsigned* __restrict__ deg,
                         long long nedges) {
    long long e = (long long)blockIdx.x * blockDim.x + threadIdx.x;
    if (e < nedges) atomicAdd(&deg[col[e]], 1u);
}

// ---------------- K4: dinv = rsqrt(deg) (deg >= 1 always) -----------------
__global__ void k_dinv(const unsigned* __restrict__ deg, float* __restrict__ dinv,
                       int n) {
    int i = blockIdx.x * blockDim.x + threadIdx.x;
    if (i < n) dinv[i] = rsqrtf((float)deg[i]);
}

// ---------------- K5: out = b + xw * dinv^2  (self-loop + bias) -----------
__global__ void k_self_init(const float* __restrict__ xw,
                            const float* __restrict__ dinv,
                            const float* __restrict__ b,
                            float* __restrict__ out, int n) {
    const int node = (int)((blockIdx.x * (unsigned)blockDim.x + threadIdx.x) >> 5);
    const int lane = threadIdx.x & 31;
    if (node >= n) return;
    const float d = dinv[node];
    const float s = d * d;
    const float4 xv = reinterpret_cast<const float4*>(xw + (size_t)node * D)[lane];
    const float4 bv = reinterpret_cast<const float4*>(b)[lane];
    float4 o;
    o.x = bv.x + xv.x * s;
    o.y = bv.y + xv.y * s;
    o.z = bv.z + xv.z * s;
    o.w = bv.w + xv.w * s;
    reinterpret_cast<float4*>(out + (size_t)node * D)[lane] = o;
}

// ---------------- K6: edge scatter (the bandwidth-dominant phase) ---------
// One wave per edge: 512B gather of xw[row] (global_load_b128 per lane),
// scale, 4x global_atomic_add_f32 per lane into out[col]. xw and out both
// fit in the 192MB L2, so the mean-degree-32 reuse is captured on-chip.
__global__ void k_scatter(const float* __restrict__ xw,
                          const float* __restrict__ dinv,
                          const int* __restrict__ row,
                          const int* __restrict__ col,
                          float* __restrict__ out, long long nedges) {
    const long long wave =
        ((long long)blockIdx.x * blockDim.x + threadIdx.x) >> 5;
    const int lane = threadIdx.x & 31;
    if (wave >= nedges) return;
    const int r = row[wave];
    const int c = col[wave];
    const float norm = dinv[r] * dinv[c];

    const float4 v = reinterpret_cast<const float4*>(xw + (size_t)r * D)[lane];
    float* o = out + (size_t)c * D + lane * 4;
    unsafeAtomicAdd(o + 0, v.x * norm);
    unsafeAtomicAdd(o + 1, v.y * norm);
    unsafeAtomicAdd(o + 2, v.z * norm);
    unsafeAtomicAdd(o + 3, v.w * norm);
}

// ---------------------------------------------------------------------------
extern "C" void kernel_launch(void* const* d_in, const int* in_sizes, int n_in,
                              void* d_out, int out_size, void* d_ws, size_t ws_size,
                              hipStream_t stream) {
    const float* x  = (const float*)d_in[0];   // [N, 128]
    const int*   ei = (const int*)d_in[1];     // [2, E]  (row = ei, col = ei + E)
    const float* W  = (const float*)d_in[3];   // [128, 128]
    const float* b  = (const float*)d_in[4];   // [128]
    float*       out = (float*)d_out;          // [N, 128]

    const int       N = in_sizes[0] / D;       // 100000
    const long long E = in_sizes[1] / 2;       // 3200000
    const int* row = ei;
    const int* col = ei + E;

    // Workspace: xw[N*D] f32 | deg[N] u32 | dinv[N] f32 | Wt[8192] v2f (~52MB)
    float*    xw   = (float*)d_ws;
    unsigned* deg  = (unsigned*)((char*)d_ws + (size_t)N * D * sizeof(float));
    float*    dinv = (float*)(deg + N);
    float*    Wt   = dinv + N;

    const int T = 256;                         // 8 waves per block

    // K0: repack W into fragment order (8192 v2f elements)
    k_pack_w<<<(NT * KK * 32 + T - 1) / T, T, 0, stream>>>(W, Wt);

    // K1: deg = 1 (self-loops)
    k_init_deg<<<(N + T - 1) / T, T, 0, stream>>>(deg, N);

    // K2: xw = x @ W   (one wave per 16-row tile)
    {
        const int ntiles = (N + 15) / 16;      // 6250 waves
        const int waves_per_block = T / 32;    // 8
        const int blocks = (ntiles + waves_per_block - 1) / waves_per_block;
        k_gemm_wmma<<<blocks, T, 0, stream>>>(x, Wt, xw, N);
    }

    // K3: histogram target degrees
    k_degree<<<(int)((E + T - 1) / T), T, 0, stream>>>(col, deg, E);

    // K4: dinv = rsqrt(deg)
    k_dinv<<<(N + T - 1) / T, T, 0, stream>>>(deg, dinv, N);

    // K5: out = b + xw * dinv^2   (one wave per node)
    {
        const long long waves = N;
        const int blocks = (int)((waves * 32 + T - 1) / T);
        k_self_init<<<blocks, T, 0, stream>>>(xw, dinv, b, out, N);
    }

    // K6: edge scatter (one wave per edge)
    {
        const long long threads = E * 32;
        const int blocks = (int)((threads + T - 1) / T);
        k_scatter<<<blocks, T, 0, stream>>>(xw, dinv, row, col, out, E);
    }
}